// FactorizedReduceSPDNet_34832184770667
// MI455X (gfx1250) — compile-verified
//
#include <hip/hip_runtime.h>
#include <math.h>

typedef __attribute__((ext_vector_type(2))) float v2f;
typedef __attribute__((ext_vector_type(8))) float v8f;

#define EPS_REEIG 1e-4f

// ---- geometry -------------------------------------------------------------
constexpr int N1  = 100;           // ni
constexpr int NP1 = 112;           // padded (7 * 16)
constexpr int LD1 = 113;           // odd LDS stride (bank-conflict free col reads)
constexpr int SZ1 = NP1 * LD1;
constexpr int MT1 = 7;
constexpr int K41 = 28;            // K steps of 4 (112/4)

constexpr int N2  = 50;            // no
constexpr int NP2 = 64;            // padded (4 * 16)
constexpr int LD2 = 65;
constexpr int SZ2 = NP2 * LD2;
constexpr int MT2 = 4;
constexpr int K42 = 16;            // 64/4

constexpr int SZW = NP1 * LD2;     // 112x64 padded W / T buffers
constexpr int NMAT = 2048;         // B*C = 128*16
constexpr int NWAVES = 8;          // 256-thread blocks, wave32

// ---- generic wave-tiled fp32 WMMA GEMM on LDS -----------------------------
// D[MT*16 x NT*16] = alpha * op(A) @ op(B) + beta * D   (K = K4*4)
// Contiguous-chunk tile ownership: each wave's tiles lie in ONE output row,
// so A fragments are loaded once per row and reused across column tiles.
// Fragments are preloaded in K-chunks (burst ds_loads, then burst wmma).
template <int TA, int TB, int MT, int NT, int K4>
__device__ __forceinline__ void wg_gemm(float* __restrict__ D,
                                        const float* __restrict__ A,
                                        const float* __restrict__ B,
                                        int ldd, int lda, int ldb,
                                        float alpha, float beta) {
  constexpr int TOT = MT * NT;
  constexpr int PER = (TOT + NWAVES - 1) / NWAVES;   // tiles per wave
  static_assert(PER <= NT && (NT % PER) == 0, "chunk must stay within a row");
  constexpr int KC = ((K4 % 8) == 0) ? 8 : (((K4 % 7) == 0) ? 7 : K4);

  const int lane = threadIdx.x & 31;
  const int wave = (int)threadIdx.x >> 5;
  const int half = lane >> 4;   // 0 or 1 -> per-lane K base (ISA A/B layout)
  const int l16  = lane & 15;

  const int t0 = wave * PER;
  if (t0 < TOT) {
    const int tm  = t0 / NT;
    const int tn0 = t0 % NT;
    const int m   = tm * 16 + l16;

    v8f acc[PER];
#pragma unroll
    for (int p = 0; p < PER; ++p)
      acc[p] = (v8f){0.f, 0.f, 0.f, 0.f, 0.f, 0.f, 0.f, 0.f};

#pragma unroll
    for (int kc = 0; kc < K4; kc += KC) {
      v2f aF[KC];
#pragma unroll
      for (int q = 0; q < KC; ++q) {
        const int k = (kc + q) * 4 + half * 2;
        if (TA) { aF[q].x = A[(k    ) * lda + m]; aF[q].y = A[(k + 1) * lda + m]; }
        else    { aF[q].x = A[m * lda + k    ];  aF[q].y = A[m * lda + k + 1]; }
      }
#pragma unroll
      for (int p = 0; p < PER; ++p) {
        const int n = (tn0 + p) * 16 + l16;
        v2f bF[KC];
#pragma unroll
        for (int q = 0; q < KC; ++q) {
          const int k = (kc + q) * 4 + half * 2;
          if (TB) { bF[q].x = B[n * ldb + k    ];  bF[q].y = B[n * ldb + k + 1]; }
          else    { bF[q].x = B[(k    ) * ldb + n]; bF[q].y = B[(k + 1) * ldb + n]; }
        }
#pragma unroll
        for (int q = 0; q < KC; ++q)
          acc[p] = __builtin_amdgcn_wmma_f32_16x16x4_f32(false, aF[q], false, bF[q],
                                                         (short)0, acc[p], false, false);
      }
    }
    // writeback
#pragma unroll
    for (int p = 0; p < PER; ++p) {
      const int n  = (tn0 + p) * 16 + l16;
      const int mb = tm * 16 + half * 8;
#pragma unroll
      for (int r = 0; r < 8; ++r) {
        float* ptr = &D[(mb + r) * ldd + n];
        float d0 = (beta != 0.f) ? *ptr : 0.f;
        *ptr = alpha * acc[p][r] + beta * d0;
      }
    }
  }
}

// ---- elementwise helpers (caller supplies barriers) -----------------------
__device__ __forceinline__ void wg_fill(float* d, float v, int n) {
  for (int i = threadIdx.x; i < n; i += blockDim.x) d[i] = v;
}
__device__ __forceinline__ void wg_scale_copy(float* d, const float* s, float a, int n) {
  for (int i = threadIdx.x; i < n; i += blockDim.x) d[i] = a * s[i];
}
__device__ __forceinline__ void wg_axpy(float* d, const float* s, float a, int n) {
  for (int i = threadIdx.x; i < n; i += blockDim.x) d[i] += a * s[i];
}
__device__ __forceinline__ void wg_add_diag(float* d, float a, int np, int ld) {
  for (int i = threadIdx.x; i < np; i += blockDim.x) d[i * ld + i] += a;
}

// load dense n x n global matrix into padded LDS (np x ld), zero pads,
// optional symmetrization and pad-diagonal value. Caller barriers after.
__device__ void wg_load_pad(float* dst, const float* __restrict__ src, int n,
                            int np, int ld, int sym, float diagPadVal) {
  for (int i = threadIdx.x; i < np * ld; i += blockDim.x) dst[i] = 0.f;
  __syncthreads();
  for (int e = threadIdx.x; e < n * n; e += blockDim.x) {
    int i = e / n, j = e % n;
    float v = src[i * n + j];
    if (sym) v = 0.5f * (v + src[j * n + i]);
    dst[i * ld + j] = v;
  }
  if (diagPadVal != 0.f)
    for (int i = n + threadIdx.x; i < np; i += blockDim.x)
      dst[i * ld + i] = diagPadVal;
}

// Frobenius norm of padded np x np matrix (stride ld). red[] has blockDim floats.
__device__ float wg_fnorm(const float* X, int np, int ld, float* red) {
  float s = 0.f;
  for (int e = threadIdx.x; e < np * np; e += blockDim.x) {
    int i = e / np, j = e % np;
    float v = X[i * ld + j];
    s += v * v;
  }
  red[threadIdx.x] = s;
  __syncthreads();
  if (threadIdx.x == 0) {
    float t = 0.f;
    for (int i = 0; i < (int)blockDim.x; ++i) t += red[i];
    red[0] = sqrtf(t);
  }
  __syncthreads();
  float r = red[0];
  __syncthreads();
  return r;
}

// Coupled Newton-Schulz:  Y -> sqrt(A)/sqrt(c),  Z -> sqrt(c)*A^{-1/2}
// Input: Y holds padded A (pad diag > 0). Pointers rotate; pass by reference.
__device__ void wg_ns_sqrt(float*& Y, float*& Z, float*& T, float*& U,
                           float* red, int niter, float& cOut) {
  float c = wg_fnorm(Y, NP2, LD2, red);
  float k = (c > 0.f) ? 1.f / c : 0.f;
  wg_scale_copy(Y, Y, k, SZ2);           // Y = A / c
  wg_fill(Z, 0.f, SZ2);
  __syncthreads();
  wg_add_diag(Z, 1.f, NP2, LD2);         // Z = I
  __syncthreads();
  for (int it = 0; it < niter; ++it) {
    wg_gemm<0, 0, MT2, MT2, K42>(T, Z, Y, LD2, LD2, LD2, -0.5f, 0.f);
    __syncthreads();
    wg_add_diag(T, 1.5f, NP2, LD2);      // T = 1.5 I - 0.5 Z Y
    __syncthreads();
    wg_gemm<0, 0, MT2, MT2, K42>(U, Y, T, LD2, LD2, LD2, 1.f, 0.f);  // Ynew
    __syncthreads();
    wg_gemm<0, 0, MT2, MT2, K42>(Y, T, Z, LD2, LD2, LD2, 1.f, 0.f);  // Znew -> old Y buf
    __syncthreads();
    float *oy = Y, *oz = Z, *ot = T, *ou = U;
    Y = ou; Z = oy; T = oz; U = ot;
  }
  cOut = c;
}

// ===========================================================================
// Kernel 1: ReEig via matrix-sign Newton-Schulz.  grid = 2048 (B*Ci)
//   Y = 0.5 * ( sym(X) + eps I + (sym(X)-eps I) @ sign(sym(X)-eps I) )
// ===========================================================================
__global__ void __launch_bounds__(256)
reeig_kernel(const float* __restrict__ x, float* __restrict__ Yout) {
  extern __shared__ float sm[];
  float* Z   = sm;
  float* Sa  = Z + SZ1;
  float* Sb  = Sa + SZ1;
  float* red = Sb + SZ1;
  const float* X  = x + (size_t)blockIdx.x * (N1 * N1);
  float*       Yg = Yout + (size_t)blockIdx.x * (N1 * N1);

  // Z = sym(X) - eps I, zero-padded
  wg_fill(Z, 0.f, SZ1);
  __syncthreads();
  for (int e = threadIdx.x; e < N1 * N1; e += blockDim.x) {
    int i = e / N1, j = e % N1;
    float v = 0.5f * (X[i * N1 + j] + X[j * N1 + i]);
    if (i == j) v -= EPS_REEIG;
    Z[i * LD1 + j] = v;
  }
  __syncthreads();
  float c = wg_fnorm(Z, NP1, LD1, red);
  float invc = (c > 0.f) ? 1.f / c : 0.f;
  wg_scale_copy(Z, Z, invc, SZ1);
  __syncthreads();

  // sign iteration: Z <- 1.5 Z - 0.5 Z^3
  for (int it = 0; it < 16; ++it) {
    wg_gemm<0, 0, MT1, MT1, K41>(Sa, Z, Z, LD1, LD1, LD1, 1.f, 0.f);    // Z^2
    __syncthreads();
    wg_gemm<0, 0, MT1, MT1, K41>(Sb, Z, Sa, LD1, LD1, LD1, -0.5f, 0.f); // -0.5 Z^3
    __syncthreads();
    wg_axpy(Sb, Z, 1.5f, SZ1);
    __syncthreads();
    float* t = Z; Z = Sb; Sb = t;
  }

  // Sa = sym(X); Sb = sym(X) @ sign(S)
  wg_fill(Sa, 0.f, SZ1);
  __syncthreads();
  for (int e = threadIdx.x; e < N1 * N1; e += blockDim.x) {
    int i = e / N1, j = e % N1;
    Sa[i * LD1 + j] = 0.5f * (X[i * N1 + j] + X[j * N1 + i]);
  }
  __syncthreads();
  wg_gemm<0, 0, MT1, MT1, K41>(Sb, Sa, Z, LD1, LD1, LD1, 1.f, 0.f);
  __syncthreads();
  for (int e = threadIdx.x; e < N1 * N1; e += blockDim.x) {
    int i = e / N1, j = e % N1;
    float r = Sa[i * LD1 + j] + Sb[i * LD1 + j] - EPS_REEIG * Z[i * LD1 + j];
    if (i == j) r += EPS_REEIG;
    Yg[e] = 0.5f * r;
  }
}

// ===========================================================================
// Kernel 2: BiMap  out[b,o] = sum_i W[o,i]^T Y[b,i] W[o,i].  grid = 2048 (b*16+o)
// ===========================================================================
__global__ void __launch_bounds__(256)
bimap_kernel(const float* __restrict__ Y, const float* __restrict__ W,
             float* __restrict__ Xf) {
  extern __shared__ float sm[];
  float* Ys = sm;          // NP1 x LD1
  float* Ws = Ys + SZ1;    // NP1 x LD2
  float* Ts = Ws + SZW;    // NP1 x LD2
  const int b = blockIdx.x >> 4;
  const int o = blockIdx.x & 15;
  const int lane = threadIdx.x & 31;
  const int wave = (int)threadIdx.x >> 5;   // 8 waves; each owns 2 of 16 tiles
  const int half = lane >> 4;
  const int l16  = lane & 15;
  // wave-owned output tiles t = 2*wave, 2*wave+1 -> one row
  const int tmO  = wave >> 1;
  const int tn0  = (wave & 1) * 2;

  v8f acc0 = {0.f, 0.f, 0.f, 0.f, 0.f, 0.f, 0.f, 0.f};
  v8f acc1 = {0.f, 0.f, 0.f, 0.f, 0.f, 0.f, 0.f, 0.f};

  // zero pads once (data region is fully overwritten every iteration)
  wg_fill(Ys, 0.f, SZ1);
  wg_fill(Ws, 0.f, SZW);
  __syncthreads();

  for (int i = 0; i < 16; ++i) {
    const float* Yg = Y + (size_t)(b * 16 + i) * (N1 * N1);
    const float* Wg = W + (size_t)(o * 16 + i) * (N1 * N2);
    __syncthreads();
    for (int e = threadIdx.x; e < N1 * N1; e += blockDim.x) {
      int r = e / N1, cc = e % N1;
      Ys[r * LD1 + cc] = Yg[e];
    }
    for (int e = threadIdx.x; e < N1 * N2; e += blockDim.x) {
      int r = e / N2, cc = e % N2;
      Ws[r * LD2 + cc] = Wg[e];
    }
    __syncthreads();
    // T = Y @ W   (112 x 64)
    wg_gemm<0, 0, MT1, MT2, K41>(Ts, Ys, Ws, LD2, LD1, LD2, 1.f, 0.f);
    __syncthreads();
    // acc += W^T @ T  (64 x 64), K = 112; A-frags cached per K-chunk
#pragma unroll
    for (int kc = 0; kc < K41; kc += 7) {
      v2f aF[7];
#pragma unroll
      for (int q = 0; q < 7; ++q) {
        const int k = (kc + q) * 4 + half * 2;
        aF[q].x = Ws[(k    ) * LD2 + tmO * 16 + l16];   // W^T
        aF[q].y = Ws[(k + 1) * LD2 + tmO * 16 + l16];
      }
#pragma unroll
      for (int p = 0; p < 2; ++p) {
        const int n = (tn0 + p) * 16 + l16;
        v2f bF[7];
#pragma unroll
        for (int q = 0; q < 7; ++q) {
          const int k = (kc + q) * 4 + half * 2;
          bF[q].x = Ts[(k    ) * LD2 + n];
          bF[q].y = Ts[(k + 1) * LD2 + n];
        }
        v8f acc = p ? acc1 : acc0;
#pragma unroll
        for (int q = 0; q < 7; ++q)
          acc = __builtin_amdgcn_wmma_f32_16x16x4_f32(false, aF[q], false, bF[q],
                                                      (short)0, acc, false, false);
        if (p) acc1 = acc; else acc0 = acc;
      }
    }
  }
  // write dense 50x50
  float* Og = Xf + (size_t)blockIdx.x * (N2 * N2);
#pragma unroll
  for (int p = 0; p < 2; ++p) {
    const int n  = (tn0 + p) * 16 + l16;
    const int mb = tmO * 16 + half * 8;
    v8f a = p ? acc1 : acc0;
    if (n < N2) {
#pragma unroll
      for (int r = 0; r < 8; ++r) {
        const int m = mb + r;
        if (m < N2) Og[m * N2 + n] = a[r];
      }
    }
  }
}

// ===========================================================================
// Kernel 3: arithmetic mean over the 2048 matrices
// ===========================================================================
__global__ void mean_kernel(const float* __restrict__ Xf, float* __restrict__ G) {
  int e = blockIdx.x * blockDim.x + threadIdx.x;
  if (e >= N2 * N2) return;
  float s = 0.f;
  for (int k = 0; k < NMAT; ++k) s += Xf[(size_t)k * (N2 * N2) + e];
  G[e] = s * (1.f / (float)NMAT);
}

// ===========================================================================
// Kernel 4: matrix sqrt + inverse sqrt (coupled Newton-Schulz).  grid = 1
// ===========================================================================
__global__ void __launch_bounds__(256)
matfn_sqrt_kernel(const float* __restrict__ A, float* __restrict__ outSq,
                  float* __restrict__ outIsq) {
  extern __shared__ float sm[];
  float* Y = sm;
  float* Z = Y + SZ2;
  float* T = Z + SZ2;
  float* U = T + SZ2;
  float* red = U + SZ2;
  wg_load_pad(Y, A, N2, NP2, LD2, 1, 1.0f);  // symmetrize, identity pad
  __syncthreads();
  float c;
  wg_ns_sqrt(Y, Z, T, U, red, 14, c);
  float sc = sqrtf(c);
  float isc = (sc > 0.f) ? 1.f / sc : 0.f;
  for (int e = threadIdx.x; e < N2 * N2; e += blockDim.x) {
    int i = e / N2, j = e % N2;
    outSq[e]  = sc * Y[i * LD2 + j];
    outIsq[e] = isc * Z[i * LD2 + j];
  }
}

// ===========================================================================
// Kernel 5: whiten + matrix log + accumulate tangent mean.  grid = 2048
//   XT = Gisq @ X @ Gisq ;  L = log(XT)  (3x NS sqrt + Mercator series)
//   GT += L * 2^3 / 2048   (float atomics)
// ===========================================================================
__global__ void __launch_bounds__(256)
log_accum_kernel(const float* __restrict__ Xf, const float* __restrict__ Gisq,
                 float* __restrict__ GTacc) {
  extern __shared__ float sm[];
  float* B0 = sm;
  float* B1 = B0 + SZ2;
  float* B2 = B1 + SZ2;
  float* B3 = B2 + SZ2;
  float* red = B3 + SZ2;

  wg_load_pad(B0, Gisq, N2, NP2, LD2, 0, 0.f);
  __syncthreads();
  wg_load_pad(B1, Xf + (size_t)blockIdx.x * (N2 * N2), N2, NP2, LD2, 0, 0.f);
  __syncthreads();
  wg_gemm<0, 0, MT2, MT2, K42>(B2, B0, B1, LD2, LD2, LD2, 1.f, 0.f);
  __syncthreads();
  wg_gemm<0, 0, MT2, MT2, K42>(B3, B2, B0, LD2, LD2, LD2, 1.f, 0.f);  // XT
  __syncthreads();
  for (int i = N2 + threadIdx.x; i < NP2; i += blockDim.x)
    B3[i * LD2 + i] = 1.f;   // identity on pad diag (log -> 0 there)
  __syncthreads();

  // inverse scaling-and-squaring: 3 square roots
  float* Y = B3; float* Z = B0; float* T = B1; float* U = B2;
  const int S = 3;
  for (int sq = 0; sq < S; ++sq) {
    float c;
    wg_ns_sqrt(Y, Z, T, U, red, 12, c);
    wg_scale_copy(Y, Y, sqrtf(c), SZ2);   // Y = A^{1/2}, input for next round
    __syncthreads();
  }

  // M = R - I ; log(R) ~= sum_{j=1..8} (-1)^{j+1} M^j / j
  wg_scale_copy(Z, Y, 1.f, SZ2);
  __syncthreads();
  wg_add_diag(Z, -1.f, NP2, LD2);
  __syncthreads();
  float* Mb = Z; float* Pw = T; float* Acc = U; float* Tmp = Y;
  wg_scale_copy(Pw, Mb, 1.f, SZ2);
  wg_scale_copy(Acc, Mb, 1.f, SZ2);
  __syncthreads();
  for (int j = 2; j <= 8; ++j) {
    wg_gemm<0, 0, MT2, MT2, K42>(Tmp, Pw, Mb, LD2, LD2, LD2, 1.f, 0.f);
    __syncthreads();
    { float* t = Pw; Pw = Tmp; Tmp = t; }
    float coef = ((j & 1) ? 1.f : -1.f) / (float)j;
    wg_axpy(Acc, Pw, coef, SZ2);
    __syncthreads();
  }

  const float sc = (float)(1 << S) / (float)NMAT;
  for (int e = threadIdx.x; e < N2 * N2; e += blockDim.x) {
    int i = e / N2, j = e % N2;
    atomicAdd(&GTacc[e], sc * Acc[i * LD2 + j]);
  }
}

// ===========================================================================
// Kernel 6: matrix exp (scaling & squaring + Taylor).  grid = 1
// ===========================================================================
__global__ void __launch_bounds__(256)
exp_kernel(const float* __restrict__ GT, float* __restrict__ E) {
  extern __shared__ float sm[];
  float* M   = sm;
  float* Pw  = M + SZ2;
  float* Acc = Pw + SZ2;
  float* Tmp = Acc + SZ2;
  const int   S2  = 4;
  const float inv = 1.f / 16.f;

  wg_load_pad(M, GT, N2, NP2, LD2, 1, 0.f);  // symmetrize; pads zero -> exp pad = I
  __syncthreads();
  wg_scale_copy(M, M, inv, SZ2);
  __syncthreads();
  wg_scale_copy(Pw, M, 1.f, SZ2);
  wg_scale_copy(Acc, M, 1.f, SZ2);
  __syncthreads();
  wg_add_diag(Acc, 1.f, NP2, LD2);           // Acc = I + M
  __syncthreads();
  for (int j = 2; j <= 8; ++j) {
    wg_gemm<0, 0, MT2, MT2, K42>(Tmp, Pw, M, LD2, LD2, LD2, 1.f / (float)j, 0.f);
    __syncthreads();
    { float* t = Pw; Pw = Tmp; Tmp = t; }    // Pw = M^j / j!
    wg_axpy(Acc, Pw, 1.f, SZ2);
    __syncthreads();
  }
  for (int s = 0; s < S2; ++s) {
    wg_gemm<0, 0, MT2, MT2, K42>(Tmp, Acc, Acc, LD2, LD2, LD2, 1.f, 0.f);
    __syncthreads();
    { float* t = Acc; Acc = Tmp; Tmp = t; }
  }
  for (int e = threadIdx.x; e < N2 * N2; e += blockDim.x) {
    int i = e / N2, j = e % N2;
    E[e] = Acc[i * LD2 + j];
  }
}

// ===========================================================================
// Kernel 7: small product D = A @ B (@ C), optional output symmetrize. grid=1
// ===========================================================================
__global__ void __launch_bounds__(256)
tri_mul_kernel(const float* __restrict__ A, const float* __restrict__ B,
               const float* __restrict__ C, float* __restrict__ D,
               int useC, int symOut) {
  extern __shared__ float sm[];
  float* Ba = sm;
  float* Bb = Ba + SZ2;
  float* Bt = Bb + SZ2;
  wg_load_pad(Ba, A, N2, NP2, LD2, 0, 0.f);
  __syncthreads();
  wg_load_pad(Bb, B, N2, NP2, LD2, 0, 0.f);
  __syncthreads();
  wg_gemm<0, 0, MT2, MT2, K42>(Bt, Ba, Bb, LD2, LD2, LD2, 1.f, 0.f);
  __syncthreads();
  const float* src = Bt;
  if (useC) {
    wg_load_pad(Ba, C, N2, NP2, LD2, 0, 0.f);
    __syncthreads();
    wg_gemm<0, 0, MT2, MT2, K42>(Bb, Bt, Ba, LD2, LD2, LD2, 1.f, 0.f);
    __syncthreads();
    src = Bb;
  }
  for (int e = threadIdx.x; e < N2 * N2; e += blockDim.x) {
    int i = e / N2, j = e % N2;
    float v = src[i * LD2 + j];
    if (symOut) v = 0.5f * (v + src[j * LD2 + i]);
    D[e] = v;
  }
}

// ===========================================================================
// Kernel 8: final  out[k] = P @ Xf[k] @ P^T   (P = Wsq @ Gisq2).  grid = 2048
// ===========================================================================
__global__ void __launch_bounds__(256)
final_kernel(const float* __restrict__ Xf, const float* __restrict__ P,
             float* __restrict__ out) {
  extern __shared__ float sm[];
  float* Pb = sm;
  float* Xb = Pb + SZ2;
  float* Tb = Xb + SZ2;
  wg_load_pad(Pb, P, N2, NP2, LD2, 0, 0.f);
  __syncthreads();
  wg_load_pad(Xb, Xf + (size_t)blockIdx.x * (N2 * N2), N2, NP2, LD2, 0, 0.f);
  __syncthreads();
  wg_gemm<0, 0, MT2, MT2, K42>(Tb, Pb, Xb, LD2, LD2, LD2, 1.f, 0.f);
  __syncthreads();
  wg_gemm<0, 1, MT2, MT2, K42>(Xb, Tb, Pb, LD2, LD2, LD2, 1.f, 0.f);  // @ P^T
  __syncthreads();
  float* Og = out + (size_t)blockIdx.x * (N2 * N2);
  for (int e = threadIdx.x; e < N2 * N2; e += blockDim.x) {
    int i = e / N2, j = e % N2;
    Og[e] = Xb[i * LD2 + j];
  }
}

__global__ void zero_kernel(float* __restrict__ p, int n) {
  int i = blockIdx.x * blockDim.x + threadIdx.x;
  if (i < n) p[i] = 0.f;
}

// ===========================================================================
extern "C" void kernel_launch(void* const* d_in, const int* in_sizes, int n_in,
                              void* d_out, int out_size, void* d_ws, size_t ws_size,
                              hipStream_t stream) {
  (void)in_sizes; (void)n_in; (void)out_size; (void)ws_size;
  const float* x  = (const float*)d_in[0];   // (128,16,100,100)
  const float* W  = (const float*)d_in[1];   // (16,16,100,50)
  const float* bn = (const float*)d_in[2];   // (50,50)
  float* out = (float*)d_out;                // (128,16,50,50)

  // workspace layout (floats)
  float* wsY     = (float*)d_ws;                      // 2048*10000
  float* wsXf    = wsY + (size_t)NMAT * N1 * N1;      // 2048*2500
  float* wsG     = wsXf + (size_t)NMAT * N2 * N2;
  float* wsGsq   = wsG + N2 * N2;
  float* wsGisq  = wsGsq + N2 * N2;
  float* wsGT    = wsGisq + N2 * N2;
  float* wsE     = wsGT + N2 * N2;
  float* wsGn    = wsE + N2 * N2;
  float* wsGnSq  = wsGn + N2 * N2;
  float* wsGnIsq = wsGnSq + N2 * N2;
  float* wsWsq   = wsGnIsq + N2 * N2;
  float* wsWisq  = wsWsq + N2 * N2;
  float* wsP     = wsWisq + N2 * N2;

  const int REEIG_LDS = (3 * SZ1 + 256) * (int)sizeof(float);   // ~153 KB (WGP has 320 KB)
  const int BIMAP_LDS = (SZ1 + 2 * SZW) * (int)sizeof(float);   // ~109 KB
  const int MATFN_LDS = (4 * SZ2 + 256) * (int)sizeof(float);   // ~68 KB
  const int TRI_LDS   = (3 * SZ2) * (int)sizeof(float);         // ~50 KB

  zero_kernel<<<(N2 * N2 + 255) / 256, 256, 0, stream>>>(wsGT, N2 * N2);

  reeig_kernel<<<NMAT, 256, REEIG_LDS, stream>>>(x, wsY);
  bimap_kernel<<<NMAT, 256, BIMAP_LDS, stream>>>(wsY, W, wsXf);
  mean_kernel<<<(N2 * N2 + 255) / 256, 256, 0, stream>>>(wsXf, wsG);
  matfn_sqrt_kernel<<<1, 256, MATFN_LDS, stream>>>(wsG, wsGsq, wsGisq);
  log_accum_kernel<<<NMAT, 256, MATFN_LDS, stream>>>(wsXf, wsGisq, wsGT);
  exp_kernel<<<1, 256, MATFN_LDS, stream>>>(wsGT, wsE);
  tri_mul_kernel<<<1, 256, TRI_LDS, stream>>>(wsGsq, wsE, wsGsq, wsGn, 1, 1);
  matfn_sqrt_kernel<<<1, 256, MATFN_LDS, stream>>>(wsGn, wsGnSq, wsGnIsq);
  matfn_sqrt_kernel<<<1, 256, MATFN_LDS, stream>>>(bn, wsWsq, wsWisq);
  tri_mul_kernel<<<1, 256, TRI_LDS, stream>>>(wsWsq, wsGnIsq, nullptr, wsP, 0, 0);
  final_kernel<<<NMAT, 256, TRI_LDS, stream>>>(wsXf, wsP, out);
}